// DynamicMaskHead_24257975288269
// MI455X (gfx1250) — compile-verified
//
#include <hip/hip_runtime.h>

typedef __attribute__((ext_vector_type(2))) float v2f;
typedef __attribute__((ext_vector_type(8))) float v8f;

#define IN_CH   8
#define NCHIN   10            // IN_CH + 2 rel-coord channels
#define CH      8
#define H       160
#define W       240
#define HW      (H * W)
#define NINST   128
#define NPARAM  169
#define RBLK    8             // logit rows produced per block (plus 1 halo)
#define NRB     (H / RBLK)    // 20 row-blocks
#define OH      (2 * H)
#define OW      (2 * W)
#define TPR     (W / 16)      // 15 WMMA tiles per logit row
#define NTILES  ((RBLK + 1) * TPR)   // 135

__device__ __forceinline__ v8f wmma_f32_4(v2f a, v2f b, v8f c) {
  return __builtin_amdgcn_wmma_f32_16x16x4_f32(false, a, false, b,
                                               (short)0, c, false, false);
}

__global__ __launch_bounds__(256) void condinst_mask_head_fused(
    const float* __restrict__ mask_feats,   // [8, 8, H, W]
    const float* __restrict__ locations,    // [HW, 2]
    const float* __restrict__ params,       // [128, 169]
    const int*   __restrict__ fpn_levels,   // [128]
    const float* __restrict__ inst_locs,    // [128, 2]
    const int*   __restrict__ im_inds,      // [128]
    float* __restrict__ out)                // [128, 1, OH, OW]
{
  __shared__ float s_p[NPARAM];
  __shared__ float s_log[(RBLK + 1) * W];

  const int inst = blockIdx.y;
  const int rb   = blockIdx.x;
  const int r0   = rb * RBLK;
  const int tid  = threadIdx.x;

  // ---- stage per-instance dynamic params into LDS ----
  if (tid < NPARAM) s_p[tid] = params[inst * NPARAM + tid];
  __syncthreads();

  const int   im      = im_inds[inst];
  const float soi_tab[5] = {64.f, 128.f, 256.f, 512.f, 1024.f};
  const float inv_soi = 1.0f / soi_tab[fpn_levels[inst]];
  const float ilx     = inst_locs[2 * inst + 0];
  const float ily     = inst_locs[2 * inst + 1];

  const int lane = tid & 31;
  const int wv   = tid >> 5;
  const int m    = lane & 15;   // A-matrix row (out-channel), B/C column (pixel)
  const int hsel = lane >> 4;   // lane-half: selects K-pair (A/B) or M-half (C)

  // ---- per-wave register weights in WMMA A layout (16x4 f32: M=lane&15,
  //      VGPR.x = K = 2*hsel, VGPR.y = K = 2*hsel+1; rows >= 8 zero-padded) ----
  v2f a00, a01, a02, a10, a11;
  {
    const bool vm = (m < CH);
    a00.x = vm ? s_p[m * NCHIN + (0 + 2 * hsel)] : 0.f;
    a00.y = vm ? s_p[m * NCHIN + (1 + 2 * hsel)] : 0.f;
    a01.x = vm ? s_p[m * NCHIN + (4 + 2 * hsel)] : 0.f;
    a01.y = vm ? s_p[m * NCHIN + (5 + 2 * hsel)] : 0.f;
    a02.x = (vm && !hsel) ? s_p[m * NCHIN + 8] : 0.f;   // K=10,11 zero pad
    a02.y = (vm && !hsel) ? s_p[m * NCHIN + 9] : 0.f;
    a10.x = vm ? s_p[80 + m * CH + (0 + 2 * hsel)] : 0.f;
    a10.y = vm ? s_p[80 + m * CH + (1 + 2 * hsel)] : 0.f;
    a11.x = vm ? s_p[80 + m * CH + (4 + 2 * hsel)] : 0.f;
    a11.y = vm ? s_p[80 + m * CH + (5 + 2 * hsel)] : 0.f;
  }
  float b0r[8], b1r[8], w2r[8];
#pragma unroll
  for (int v = 0; v < 8; ++v) {
    w2r[v] = s_p[144 + v];
    b0r[v] = s_p[152 + v];
    b1r[v] = s_p[160 + v];
  }
  const float b2 = s_p[168];

  const float* __restrict__ fbase = mask_feats + (size_t)im * IN_CH * HW;

  // ---- phase 1: compute 9 logit rows (8 + 1 halo) into LDS ----
  for (int t = wv; t < NTILES; t += 8) {
    const int trow  = t / TPR;
    const int ctile = t % TPR;
    const int gr    = min(r0 + trow, H - 1);   // clamp only matters for unused halo
    const int col   = ctile * 16 + m;
    const int pix   = gr * W + col;

    // this lane supplies 2 K-rows per B chunk; feat channels needed:
    //   hsel==0: {f2,f3},{f6,f7} (+ rel coords)   hsel==1: {f0,f1},{f4,f5}
    const int cA = hsel ? 0 : 2;
    const int cB = hsel ? 4 : 6;
    const float fA0 = fbase[(size_t)(cA + 0) * HW + pix];
    const float fA1 = fbase[(size_t)(cA + 1) * HW + pix];
    const float fB0 = fbase[(size_t)(cB + 0) * HW + pix];
    const float fB1 = fbase[(size_t)(cB + 1) * HW + pix];

    float rx = 0.f, ry = 0.f;
    if (!hsel) {
      rx = (ilx - locations[2 * pix + 0]) * inv_soi;
      ry = (ily - locations[2 * pix + 1]) * inv_soi;
    }
    // B chunks (4x16 f32: lanes 0-15 K={0,1}, lanes 16-31 K={2,3} of the chunk)
    v2f bc0, bc1, bc2;
    bc0.x = hsel ? fA0 : rx;   bc0.y = hsel ? fA1 : ry;    // K 0..3: relx,rely,f0,f1
    bc1.x = hsel ? fB0 : fA0;  bc1.y = hsel ? fB1 : fA1;   // K 4..7: f2..f5
    bc2.x = hsel ? 0.f : fB0;  bc2.y = hsel ? 0.f : fB1;   // K 8..11: f6,f7,0,0

    // layer 0: D[16x16] = W0_pad[16x12] x X[12x16]
    v8f acc = {};
    acc = wmma_f32_4(a00, bc0, acc);
    acc = wmma_f32_4(a01, bc1, acc);
    acc = wmma_f32_4(a02, bc2, acc);
#pragma unroll
    for (int v = 0; v < 8; ++v)            // C layout: VGPR v <-> M = v + 8*hsel
      acc[v] = fmaxf(acc[v] + (hsel ? 0.f : b0r[v]), 0.f);

    // relayout C -> B for layer 1: X1[c][n] lives in lane n (n<16), VGPR c
    float xs[8];
#pragma unroll
    for (int v = 0; v < 8; ++v) xs[v] = __shfl(acc[v], m, 32);
    v2f d0, d1;
    d0.x = hsel ? xs[2] : xs[0];  d0.y = hsel ? xs[3] : xs[1];
    d1.x = hsel ? xs[6] : xs[4];  d1.y = hsel ? xs[7] : xs[5];

    // layer 1: 8x8 (K padded to 8, two K=4 steps)
    v8f acc1 = {};
    acc1 = wmma_f32_4(a10, d0, acc1);
    acc1 = wmma_f32_4(a11, d1, acc1);
#pragma unroll
    for (int v = 0; v < 8; ++v)
      acc1[v] = fmaxf(acc1[v] + (hsel ? 0.f : b1r[v]), 0.f);

    // layer 2: 1x8 dot per pixel (valid data sits in lanes 0-15)
    if (!hsel) {
      float lg = b2;
#pragma unroll
      for (int v = 0; v < 8; ++v) lg = fmaf(w2r[v], acc1[v], lg);
      s_log[trow * W + col] = lg;
    }
  }
  __syncthreads();

  // ---- phase 2: aligned_bilinear x2 from LDS, non-temporal stores ----
  // out[i][j] samples logit rows y0=max(i-1,0)>>1, y1=min(y0+1,H-1),
  // fy=(max(i-1,0)&1)*0.5 (same in x). All accessed rows lie in [r0, r0+RBLK].
  const int iLo   = (rb == 0) ? 0 : (2 * r0 + 1);
  const int iHi   = min(2 * r0 + 2 * RBLK, OH - 1);
  const int total = (iHi - iLo + 1) * OW;
  float* __restrict__ outp = out + (size_t)inst * OH * OW;

  for (int idx = tid; idx < total; idx += 256) {
    const int i  = iLo + idx / OW;
    const int j  = idx % OW;
    const int iy = max(i - 1, 0);
    const int jx = max(j - 1, 0);
    const int y0 = iy >> 1;
    const int y1 = min(y0 + 1, H - 1);
    const int x0 = jx >> 1;
    const int x1 = min(x0 + 1, W - 1);
    const float fy = (iy & 1) ? 0.5f : 0.0f;
    const float fx = (jx & 1) ? 0.5f : 0.0f;
    const float* rA = s_log + (y0 - r0) * W;
    const float* rB = s_log + (y1 - r0) * W;
    const float top = rA[x0] * (1.f - fx) + rA[x1] * fx;
    const float bot = rB[x0] * (1.f - fx) + rB[x1] * fx;
    const float val = top * (1.f - fy) + bot * fy;
    __builtin_nontemporal_store(val, &outp[(size_t)i * OW + j]);
  }
}

extern "C" void kernel_launch(void* const* d_in, const int* in_sizes, int n_in,
                              void* d_out, int out_size, void* d_ws, size_t ws_size,
                              hipStream_t stream) {
  const float* mask_feats = (const float*)d_in[0];
  const float* locations  = (const float*)d_in[1];
  const float* params     = (const float*)d_in[2];
  const int*   fpn        = (const int*)d_in[3];
  const float* ilocs      = (const float*)d_in[4];
  const int*   iminds     = (const int*)d_in[5];
  float* outp = (float*)d_out;

  dim3 grid(NRB, NINST);   // 20 row-blocks x 128 instances
  condinst_mask_head_fused<<<grid, 256, 0, stream>>>(
      mask_feats, locations, params, fpn, ilocs, iminds, outp);
}